// ts_zscore_14362370638093
// MI455X (gfx1250) — compile-verified
//
#include <hip/hip_runtime.h>
#include <stdint.h>

// Rolling z-score, WINDOW=20, over (B=64, T=2048, F=128) fp32; output (B, F, s=2029).
// Memory-bound: ~134 MB traffic -> ~5.7us floor at 23.3 TB/s. Strategy:
//   - TDM tensor_load_to_lds: one DMA descriptor stages the 83x128 input tile
//     into LDS (TENSORcnt path, hardware OOB zero-fill for the tail chunk)
//   - per-thread sliding sum/sum^2 (O(1) per output)
//   - async LDS->global stores (ASYNCcnt path) for the coalesced transposed
//     write-out; no VGPR round trip.

#define WINDOW  20
#define EPS_    1e-4f
#define B_      64
#define T_      2048
#define F_      128
#define S_      (T_ - WINDOW + 1)          // 2029
#define CHUNK   64                         // outputs (t') per block
#define TTILE   (CHUNK + WINDOW - 1)       // 83 input timesteps per block
#define NCHUNKS ((S_ + CHUNK - 1) / CHUNK) // 32

typedef unsigned int v4u __attribute__((ext_vector_type(4)));
typedef int          v4i __attribute__((ext_vector_type(4)));
typedef int          v8i __attribute__((ext_vector_type(8)));

__global__ __launch_bounds__(128)
void ts_zscore_14362370638093_kernel(const float* __restrict__ in,
                                     float* __restrict__ out) {
    __shared__ __align__(16) float lds_in[TTILE * F_];        // 42,496 B
    __shared__ __align__(16) float lds_out[F_ * (CHUNK + 1)]; // 33,280 B, pad 65 vs banks

    const int tid = threadIdx.x;          // 0..127
    const int b   = blockIdx.y;           // 0..63
    const int t0  = blockIdx.x * CHUNK;   // chunk start in t'
    const int valid = (S_ - t0 < CHUNK) ? (S_ - t0) : CHUNK;

    // ---- Phase 1: TDM DMA of input tile [t0, t0+TTILE) x F_ into LDS ----
    // Wave 0 only: TDM ignores EXEC, so issuing from every wave would move the
    // tile 4x. tensor_dim1 = T-t0 gives HW zero-fill past the end of the batch
    // slice, so the tail chunk needs no clamping.
    if (tid < 32) {
        const uint64_t gaddr =
            (uint64_t)(uintptr_t)(in + (size_t)b * T_ * F_ + (size_t)t0 * F_);
        const uint32_t laddr = (uint32_t)(uintptr_t)(void*)lds_in; // low 32b == LDS offset

        // D# group 0: count=1 | lds_addr | global_addr[56:0] | type=2
        v4u g0;
        g0.x = 1u;
        g0.y = laddr;
        g0.z = (uint32_t)gaddr;
        g0.w = (uint32_t)(gaddr >> 32) | (2u << 30);

        // D# group 1: data_size=2 (4B), tensor_dim0=128, tensor_dim1=T-t0,
        // tile_dim0=128, tile_dim1=TTILE, tile_dim2=0, dim0_stride=128.
        const uint32_t td0 = (uint32_t)F_;
        const uint32_t td1 = (uint32_t)(T_ - t0);
        v8i g1;
        g1[0] = (int)(2u << 16);                                     // data_size
        g1[1] = (int)((td0 & 0xFFFFu) << 16);                        // dim0 lo16
        g1[2] = (int)((td0 >> 16) | ((td1 & 0xFFFFu) << 16));        // dim0 hi / dim1 lo
        g1[3] = (int)(((td1 >> 16) & 0xFFFFu) | ((uint32_t)F_ << 16)); // dim1 hi / tile_dim0
        g1[4] = (int)TTILE;                                          // tile_dim1, tile_dim2=0
        g1[5] = (int)F_;                                             // dim0_stride lo32
        g1[6] = 0;                                                   // stride hi / dim1_stride lo
        g1[7] = 0;                                                   // dim1_stride hi

        v4i gz = {0, 0, 0, 0};
#if defined(__clang_major__) && (__clang_major__ >= 23)
        v8i gz8 = {0, 0, 0, 0, 0, 0, 0, 0};
        __builtin_amdgcn_tensor_load_to_lds(g0, g1, gz, gz, gz8, 0);
#else
        __builtin_amdgcn_tensor_load_to_lds(g0, g1, gz, gz, 0);
#endif
        __builtin_amdgcn_s_wait_tensorcnt(0);
    }
    __syncthreads();

    // ---- Phase 2: per-feature sliding window mean/var -> LDS tile ----
    const int f = tid;
    float sum = 0.0f, sq = 0.0f;
#pragma unroll
    for (int k = 0; k < WINDOW - 1; ++k) {
        float v = lds_in[k * F_ + f];
        sum += v;
        sq  += v * v;
    }
    for (int i = 0; i < valid; ++i) {
        float vn = lds_in[(i + WINDOW - 1) * F_ + f];
        sum += vn;
        sq  += vn * vn;
        float mean = sum * (1.0f / WINDOW);
        float var  = (sq - sum * mean) * (1.0f / (WINDOW - 1));
        var = fmaxf(var, 0.0f);
        float sd = sqrtf(var);
        lds_out[f * (CHUNK + 1) + i] = (vn - mean) / (sd + EPS_);
        float vo = lds_in[i * F_ + f];
        sum -= vo;
        sq  -= vo * vo;
    }
    __syncthreads();

    // ---- Phase 3: async LDS->global transposed write-out ----
    // 128 threads = 2 rows x 64 cols per iteration; lanes 0..31 hit 32
    // consecutive t' of one f-row -> coalesced. s=2029 odd => rows not 16B
    // aligned => B32 stores.
    const uint64_t obase = (uint64_t)(uintptr_t)out;
    const uint32_t lob   = (uint32_t)(uintptr_t)(void*)lds_out;
    const int col = tid & 63;
    const int sub = tid >> 6;
    if (valid == CHUNK) {
#pragma unroll 4
        for (int r = 0; r < F_; r += 2) {
            int row = r + sub;
            uint64_t ga = obase +
                (((uint64_t)(b * F_ + row) * (uint64_t)S_ + (uint64_t)(t0 + col)) << 2);
            uint32_t la = lob + (uint32_t)(row * (CHUNK + 1) + col) * 4u;
            asm volatile("global_store_async_from_lds_b32 %0, %1, off"
                         :: "v"(ga), "v"(la) : "memory");
        }
    } else {
        for (int r = 0; r < F_; r += 2) {
            int row = r + sub;
            if (col < valid) {
                uint64_t ga = obase +
                    (((uint64_t)(b * F_ + row) * (uint64_t)S_ + (uint64_t)(t0 + col)) << 2);
                uint32_t la = lob + (uint32_t)(row * (CHUNK + 1) + col) * 4u;
                asm volatile("global_store_async_from_lds_b32 %0, %1, off"
                             :: "v"(ga), "v"(la) : "memory");
            }
        }
    }
    asm volatile("s_wait_asynccnt 0" ::: "memory");
}

extern "C" void kernel_launch(void* const* d_in, const int* in_sizes, int n_in,
                              void* d_out, int out_size, void* d_ws, size_t ws_size,
                              hipStream_t stream) {
    (void)in_sizes; (void)n_in; (void)d_ws; (void)ws_size; (void)out_size;
    const float* in = (const float*)d_in[0];
    float* out      = (float*)d_out;
    dim3 grid(NCHUNKS, B_, 1);
    ts_zscore_14362370638093_kernel<<<grid, 128, 0, stream>>>(in, out);
}